// AdditiveAttention_816043786499
// MI455X (gfx1250) — compile-verified
//
#include <hip/hip_runtime.h>
#include <math.h>

// Problem constants (from the reference)
constexpr int CB  = 4;    // batch
constexpr int QN  = 512;  // queries
constexpr int KVN = 512;  // keys/values
constexpr int QD  = 512;  // q/k feature dim
constexpr int CH  = 256;  // hidden dim H
constexpr int VD  = 256;  // value dim

typedef __attribute__((ext_vector_type(2))) float v2f;
typedef __attribute__((ext_vector_type(8))) float v8f;

// ---------------------------------------------------------------------------
// Hardware tanh (gfx1250 TRANS op). Trailing v_nop satisfies the CDNA5 rule:
// one independent op or nop between a trans op and any use of its result.
// ---------------------------------------------------------------------------
__device__ __forceinline__ float htanh(float x) {
  float y;
  asm("v_tanh_f32 %0, %1\n\tv_nop" : "=v"(y) : "v"(x));
  return y;
}

// ---------------------------------------------------------------------------
// Kernel 1: qh = q @ Wq^T  and  kh = k @ Wk^T   (fp32 WMMA 16x16x4)
// One wave per 16(row) x 16(h) output tile.
// A (16x4 f32): lane<16 holds M=lane, K=0,1 in v0,v1; lane>=16 holds K=2,3.
// B (4x16 f32): lane<16 holds N=lane, K=0,1; lane>=16 holds N=lane-16, K=2,3.
// C/D (16x16 f32): vgpr i: lanes0-15 -> M=i, lanes16-31 -> M=i+8; N=lane&15.
// ---------------------------------------------------------------------------
__global__ __launch_bounds__(32) void proj_kernel(
    const float* __restrict__ q,  const float* __restrict__ Wq,
    const float* __restrict__ k,  const float* __restrict__ Wk,
    float* __restrict__ qh, float* __restrict__ kh) {
  int tile = blockIdx.x;
  int ht = tile & 15;          // CH/16 = 16 h-tiles
  int rt = tile >> 4;          // 0..255 row tiles (128 for q, 128 for k)
  const float* X; const float* W; float* O;
  constexpr int QTILES = (CB * QN) / 16;   // 128
  if (rt < QTILES) { X = q; W = Wq; O = qh; }
  else             { X = k; W = Wk; O = kh; rt -= QTILES; }

  const int lane = threadIdx.x;
  const int r    = lane & 15;
  const int koff = (lane >> 4) << 1;       // 0 or 2
  const int m0 = rt * 16, h0 = ht * 16;

  const float* arow = X + (size_t)(m0 + r) * QD;
  const float* brow = W + (size_t)(h0 + r) * QD;

  v8f c = {0.f, 0.f, 0.f, 0.f, 0.f, 0.f, 0.f, 0.f};
  for (int kk = 0; kk < QD; kk += 4) {
    v2f a, b;
    a.x = arow[kk + koff];     a.y = arow[kk + koff + 1];
    b.x = brow[kk + koff];     b.y = brow[kk + koff + 1];
    c = __builtin_amdgcn_wmma_f32_16x16x4_f32(false, a, false, b,
                                              (short)0, c, false, false);
  }
  const int mbase = m0 + ((lane >> 4) << 3);
  #pragma unroll
  for (int i = 0; i < 8; ++i)
    O[(size_t)(mbase + i) * CH + h0 + r] = c[i];
}

// ---------------------------------------------------------------------------
// Kernel 2: scores[b,q,k] = sum_h wv[h] * tanh(qh[b,q,h] + kh[b,k,h])
// Fused — never materializes the [B,QN,KVN,H] energy tensor (1 GB!).
// Block = 256 threads handles a 16x16 (q,k) tile; qh/kh tiles staged in LDS
// (PAD=260 kills the stride-256 bank conflicts), 16x reuse per element.
// ---------------------------------------------------------------------------
__global__ __launch_bounds__(256) void score_kernel(
    const float* __restrict__ qh, const float* __restrict__ kh,
    const float* __restrict__ wv, float* __restrict__ scores) {
  constexpr int PAD = CH + 4;  // 260 dwords: (260 % 64)=4 -> conflict-free
  __shared__ __align__(16) float qs[16 * PAD];
  __shared__ __align__(16) float ks[16 * PAD];
  __shared__ __align__(16) float ws[CH];

  int tile = blockIdx.x;                 // b*(32*32) + qt*32 + kt
  int kt = tile & 31;
  int qt = (tile >> 5) & 31;
  int b  = tile >> 10;
  int q0 = qt * 16, k0 = kt * 16;
  int tid = threadIdx.x;

  const float* qg = qh + ((size_t)b * QN  + q0) * CH;
  const float* kg = kh + ((size_t)b * KVN + k0) * CH;
  for (int i = tid; i < 16 * CH; i += 256) {
    int rr = i >> 8;            // /CH (CH==256)
    int cc = i & (CH - 1);
    qs[rr * PAD + cc] = qg[(size_t)rr * CH + cc];
    ks[rr * PAD + cc] = kg[(size_t)rr * CH + cc];
  }
  if (tid < CH) ws[tid] = wv[tid];
  __syncthreads();

  const int ql = tid >> 4, kl = tid & 15;
  const float* qp = &qs[ql * PAD];
  const float* kp = &ks[kl * PAD];
  float acc = 0.f;
  #pragma unroll 4
  for (int h = 0; h < CH; h += 4) {
    float4 qv = *(const float4*)(qp + h);
    float4 kv = *(const float4*)(kp + h);
    float4 wq = *(const float4*)(&ws[h]);
    acc = fmaf(wq.x, htanh(qv.x + kv.x), acc);
    acc = fmaf(wq.y, htanh(qv.y + kv.y), acc);
    acc = fmaf(wq.z, htanh(qv.z + kv.z), acc);
    acc = fmaf(wq.w, htanh(qv.w + kv.w), acc);
  }
  scores[((size_t)b * QN + q0 + ql) * KVN + k0 + kl] = acc;
}

// ---------------------------------------------------------------------------
// Kernel 3: in-place softmax over k (512) per (b,q) row. 256 threads/row.
// ---------------------------------------------------------------------------
__global__ __launch_bounds__(256) void softmax_kernel(float* __restrict__ att) {
  __shared__ float red[256];
  float* p = att + (size_t)blockIdx.x * KVN;
  int t = threadIdx.x;
  float x0 = p[t], x1 = p[t + 256];

  red[t] = fmaxf(x0, x1);
  __syncthreads();
  for (int s = 128; s > 0; s >>= 1) {
    if (t < s) red[t] = fmaxf(red[t], red[t + s]);
    __syncthreads();
  }
  float m = red[0];
  __syncthreads();

  float e0 = __expf(x0 - m), e1 = __expf(x1 - m);
  red[t] = e0 + e1;
  __syncthreads();
  for (int s = 128; s > 0; s >>= 1) {
    if (t < s) red[t] += red[t + s];
    __syncthreads();
  }
  float inv = 1.0f / red[0];
  p[t] = e0 * inv;
  p[t + 256] = e1 * inv;
}

// ---------------------------------------------------------------------------
// Kernel 4: out[b,q,v] = attention[b,q,:] @ v[b,:,v]   (fp32 WMMA 16x16x4)
// One wave per 16x16 output tile, K-loop over KVN.
// ---------------------------------------------------------------------------
__global__ __launch_bounds__(32) void av_kernel(
    const float* __restrict__ att, const float* __restrict__ v,
    float* __restrict__ out) {
  int tile = blockIdx.x;           // b*(32*16) + qt*16 + vt
  int vt = tile & 15;              // VD/16 = 16
  int qt = (tile >> 4) & 31;       // QN/16 = 32
  int b  = tile >> 9;

  const float* A = att + (size_t)b * QN * KVN;
  const float* V = v   + (size_t)b * KVN * VD;
  float* O       = out + (size_t)b * QN * VD;

  const int lane = threadIdx.x;
  const int r    = lane & 15;
  const int koff = (lane >> 4) << 1;
  const int q0 = qt * 16, v0 = vt * 16;

  const float* arow = A + (size_t)(q0 + r) * KVN;

  v8f c = {0.f, 0.f, 0.f, 0.f, 0.f, 0.f, 0.f, 0.f};
  for (int kk = 0; kk < KVN; kk += 4) {
    v2f a, bm;
    a.x  = arow[kk + koff];
    a.y  = arow[kk + koff + 1];
    bm.x = V[(size_t)(kk + koff)     * VD + v0 + r];
    bm.y = V[(size_t)(kk + koff + 1) * VD + v0 + r];
    c = __builtin_amdgcn_wmma_f32_16x16x4_f32(false, a, false, bm,
                                              (short)0, c, false, false);
  }
  const int mbase = q0 + ((lane >> 4) << 3);
  #pragma unroll
  for (int i = 0; i < 8; ++i)
    O[(size_t)(mbase + i) * VD + v0 + r] = c[i];
}

// ---------------------------------------------------------------------------
extern "C" void kernel_launch(void* const* d_in, const int* in_sizes, int n_in,
                              void* d_out, int out_size, void* d_ws, size_t ws_size,
                              hipStream_t stream) {
  (void)in_sizes; (void)n_in; (void)out_size; (void)ws_size;
  const float* q  = (const float*)d_in[0];
  const float* k  = (const float*)d_in[1];
  const float* v  = (const float*)d_in[2];
  const float* Wq = (const float*)d_in[3];
  const float* Wk = (const float*)d_in[4];
  const float* wv = (const float*)d_in[5];

  float* out = (float*)d_out;                       // [B,QN,VD]
  float* att = out + (size_t)CB * QN * VD;          // [B,QN,KVN]

  float* qh = (float*)d_ws;                         // [B,QN,CH]  (2 MB)
  float* kh = qh + (size_t)CB * QN * CH;            // [B,KVN,CH] (2 MB)

  // 1) projections: (128+128) row-tiles x 16 h-tiles = 4096 waves
  proj_kernel<<<4096, 32, 0, stream>>>(q, Wq, k, Wk, qh, kh);
  // 2) fused add+tanh+wv reduction: 4*32*32 = 4096 tiles
  score_kernel<<<4096, 256, 0, stream>>>(qh, kh, wv, att);
  // 3) softmax over 512 keys, 4*512 = 2048 rows
  softmax_kernel<<<2048, 256, 0, stream>>>(att);
  // 4) attention @ v : 4*32*16 = 2048 tiles
  av_kernel<<<2048, 32, 0, stream>>>(att, v, out);
}